// KGFlexTFModel_35158602285715
// MI455X (gfx1250) — compile-verified
//
#include <hip/hip_runtime.h>
#include <hip/hip_bf16.h>

// ---------------------------------------------------------------------------
// KGFlex fused scoring kernel for MI455X (gfx1250, wave32).
//
// x[b] = sum_j K[u, f_j] * ( H[u,:] . G[f_j,:] + F_B[f_j] ) + I_B[item]
//        with f_j = C[u, item, j],  j = 0..FPP-1
//
// Roofline: fusing the gather avoids materializing z_u = H @ G.T (328 MB);
// total traffic ~7 MB -> sub-microsecond at 23.3 TB/s. Kernel is gather-
// latency bound, so the body is kept branch-free with all loads hoistable.
//
// Matrix path: the 40960 length-10 dots are diagonals of chained
// V_WMMA_F32_16X16X4_F32 ops. Per 16-batch tile (one wave32):
//   A[16x12] = H rows (FAC=10 zero-padded to 12, 3 chunks of K=4)
//   B[12x16] = gathered G rows pre-scaled by K[u,f] (folds k_u * (...))
// All 20 slots x 3 chunks accumulate into ONE C/D octet; the diagonal is
// extracted once at the end. Zero-padding is done with a multiply mask
// instead of guarded loads so EXEC never diverges in the hot loop.
// ---------------------------------------------------------------------------

#define U_   2000
#define NI_  5000
#define NF_  40000
#define FAC_ 10
#define FPP_ 20

typedef float v2f __attribute__((ext_vector_type(2)));
typedef float v8f __attribute__((ext_vector_type(8)));
typedef int   v4i __attribute__((ext_vector_type(4)));

__global__ __launch_bounds__(256) void kgflex_fused_wmma(
    const int*   __restrict__ user,   // [B]
    const int*   __restrict__ item,   // [B]
    const float* __restrict__ H,      // [U, FAC]
    const float* __restrict__ G,      // [NF, FAC]
    const float* __restrict__ K,      // [U, NF]
    const float* __restrict__ F_B,    // [NF]
    const float* __restrict__ I_B,    // [NI]
    const int*   __restrict__ C,      // [U, NI, FPP]
    float*       __restrict__ out,    // [B]
    int numTiles)                     // B / 16
{
    const int lane  = threadIdx.x & 31;
    const int wave  = threadIdx.x >> 5;
    const int tile  = blockIdx.x * (blockDim.x >> 5) + wave;
    if (tile >= numTiles) return;     // uniform per wave: EXEC stays all-1s

    const int n     = lane & 15;      // A-row m / B-column n owned by this lane
    const int khalf = lane >> 4;      // 0 -> K pair {0,1}, 1 -> K pair {2,3}
    const int b     = tile * 16 + n;  // batch element this lane's metadata maps to

    const int u  = user[b];
    const int it = item[b];

    // Zero-pad mask for K-chunk 2 (global K = 8..11; only {8,9} exist).
    // Both lane halves load elements {8,9} (always in-bounds); khalf==1
    // lanes multiply by 0 instead of taking a divergent-load path.
    const float mask2 = (khalf == 0) ? 1.f : 0.f;

    // --- A matrix: 3 K-chunks of V_WMMA_F32_16X16X4_F32.
    // ISA layout (16x4 f32 A): lanes 0-15 hold K={0,1} in comps {0,1};
    // lanes 16-31 hold K={2,3}. Rows are 40 B -> 8-byte aligned b64 loads.
    const float* hrow = H + (long)u * FAC_;
    v2f a0 = *(const v2f*)(hrow + 2 * khalf);        // {0,1} / {2,3}
    v2f a1 = *(const v2f*)(hrow + 4 + 2 * khalf);    // {4,5} / {6,7}
    v2f a2 = *(const v2f*)(hrow + 8);                // {8,9} (both halves)
    a2.x *= mask2;  a2.y *= mask2;

    // --- Preload this row's 20 feature indices: 5 x b128 (80-byte aligned).
    const int* crow  = C + ((long)u * NI_ + it) * FPP_;
    const v4i* crow4 = (const v4i*)crow;
    v4i cidx[5];
#pragma unroll
    for (int q = 0; q < 5; ++q) cidx[q] = crow4[q];

    const float* krow = K + (long)u * NF_;

    v8f   acc = {};     // one chained accumulator for all 60 WMMAs
    float sB  = 0.f;    // sum_j K[u,f_j] * F_B[f_j]

#pragma unroll
    for (int q = 0; q < 5; ++q) {
#pragma unroll
        for (int r = 0; r < 4; ++r) {
            const int   f   = cidx[q][r];
            const float kw  = krow[f];
            sB += kw * F_B[f];
            const float kwm = kw * mask2;   // 0 for the padded K half

            // B matrix column n = gathered G row, pre-scaled by kw.
            // Assumed 4x16 f32 B layout mirrors A: lanes 0-15 hold K-rows
            // {0,1}, lanes 16-31 K-rows {2,3}.
            const float* grow = G + (long)f * FAC_;
            v2f g0 = *(const v2f*)(grow + 2 * khalf);
            v2f g1 = *(const v2f*)(grow + 4 + 2 * khalf);
            v2f g2 = *(const v2f*)(grow + 8);
            v2f b0; b0.x = kw  * g0.x; b0.y = kw  * g0.y;
            v2f b1; b1.x = kw  * g1.x; b1.y = kw  * g1.y;
            v2f b2; b2.x = kwm * g2.x; b2.y = kwm * g2.y;

            // D[i][k] += H[b_i] . (kw_k * G[f_{k,j}]) — diagonal is wanted.
            acc = __builtin_amdgcn_wmma_f32_16x16x4_f32(false, a0, false, b0,
                                                        (short)0, acc, false, false);
            acc = __builtin_amdgcn_wmma_f32_16x16x4_f32(false, a1, false, b1,
                                                        (short)0, acc, false, false);
            acc = __builtin_amdgcn_wmma_f32_16x16x4_f32(false, a2, false, b2,
                                                        (short)0, acc, false, false);
        }
    }

    // --- Diagonal extraction.
    // C/D 16x16 f32 layout: (vgpr v, lane l<16)  -> M=v,   N=l
    //                       (vgpr v, lane l>=16) -> M=v+8, N=l-16
    // diag(i), i=0..7  : vgpr i,   lane i
    // diag(i), i=8..15 : vgpr i-8, lane i+16
    int ci = ((lane < 16) ? lane : (lane - 24)) & 7;   // comp holding my diag
    float d0 = (ci & 1) ? acc[1] : acc[0];
    float d1 = (ci & 1) ? acc[3] : acc[2];
    float d2 = (ci & 1) ? acc[5] : acc[4];
    float d3 = (ci & 1) ? acc[7] : acc[6];
    float e0 = (ci & 2) ? d1 : d0;
    float e1 = (ci & 2) ? d3 : d2;
    float diag = (ci & 4) ? e1 : e0;

    // Lane r (0..15) fetches diag(r): source lane r for r<8, lane r+16 else.
    const int src = (n < 8) ? n : (n + 16);
    const float dn = __shfl(diag, src, 32);

    if (lane < 16) {
        out[b] = dn + sB + I_B[it];
    }
}

extern "C" void kernel_launch(void* const* d_in, const int* in_sizes, int n_in,
                              void* d_out, int out_size, void* d_ws, size_t ws_size,
                              hipStream_t stream) {
    (void)n_in; (void)d_ws; (void)ws_size;
    const int*   user = (const int*)  d_in[0];
    const int*   item = (const int*)  d_in[1];
    const float* H    = (const float*)d_in[2];
    const float* G    = (const float*)d_in[3];
    const float* K    = (const float*)d_in[4];
    const float* F_B  = (const float*)d_in[5];
    const float* I_B  = (const float*)d_in[6];
    const int*   C    = (const int*)  d_in[7];
    float*       out  = (float*)d_out;

    const int B        = in_sizes[0];          // 2048
    const int numTiles = B / 16;               // 128 (16 batch rows per wave32)
    const int wavesPerBlock = 8;               // 256 threads = 8 wave32
    const int blocks   = (numTiles + wavesPerBlock - 1) / wavesPerBlock;  // 16

    kgflex_fused_wmma<<<blocks, wavesPerBlock * 32, 0, stream>>>(
        user, item, H, G, K, F_B, I_B, C, out, numTiles);
    (void)out_size;
}